// ScaledDotProductAttention_76733885710388
// MI455X (gfx1250) — compile-verified
//
#include <hip/hip_runtime.h>

// Segment (per-graph) softmax attention for MI455X (gfx1250, wave32).
//
// Inputs (setup_inputs order):
//   d_in[0] global_query  f32 [16, 64, 64]
//   d_in[1] local_key     f32 [16, 32768, 64]
//   d_in[2] local_value   f32 [16, 32768, 64]
//   d_in[3] batch         int (sorted segment ids, 0..63)  -- cast as int32
// Output (tuple, concatenated):
//   d_out[0            .. 65535   ] output f32 [16,64,64]
//   d_out[65536        .. +16*64*32768] attn f32 [16,64,32768]
//
// Design: batch is sorted -> each graph owns a contiguous edge range.
// Tile the 64 rows into 4 tiles of 16 (= WMMA M). Each (h, btile, chunk)
// block runs V_WMMA_F32_16X16X4_F32 over only the union edge range of its
// 16 rows (avg 8K edges): ~2 GFLOP total, fully hidden under the ~400MB
// mandatory HBM traffic (~17us at 23.3 TB/s).

#define H 16
#define BSZ 64
#define DK 64
#define E 32768
#define BT 16      // rows per b-tile (WMMA M)
#define NBT 4      // 64/16
#define CHUNKS 8   // edge-range split per (h,btile) for occupancy
#define NW 8       // waves per block
#define THREADS 256

typedef float v2f __attribute__((ext_vector_type(2)));
typedef float v4f __attribute__((ext_vector_type(4)));
typedef float v8f __attribute__((ext_vector_type(8)));

#define NEG_INF (-__builtin_inff())

// ---------------- workspace layout (bytes) ----------------
// [0      , 1024 )  int   bounds[65]
// [1024   , 66560)  float ml[H][NBT][CHUNKS][BT][2]   (m, l partials)
// [66560  , 70656)  float shift[H*BSZ]
// [70656  , +2MB )  float Opart[H][NBT][CHUNKS][BT][DK]
#define WS_ML     1024
#define WS_SHIFT  66560
#define WS_OPART  70656

// ---------------------------------------------------------------------------
__global__ void gat_zero_attn(float* __restrict__ attn, long n4) {
  long i = (long)blockIdx.x * blockDim.x + threadIdx.x;
  long stride = (long)gridDim.x * blockDim.x;
  v4f z = {0.f, 0.f, 0.f, 0.f};
  for (long idx = i; idx < n4; idx += stride)
    __builtin_nontemporal_store(z, ((v4f*)attn) + idx);
}

// bounds[b] = first edge index with batch[e] >= b;  bounds[64] = E
__global__ void gat_bounds(const int* __restrict__ batch, int* __restrict__ bounds) {
  int b = threadIdx.x;
  if (b > BSZ) return;
  int lo = 0, hi = E;
  while (lo < hi) {
    int mid = (lo + hi) >> 1;
    if (batch[mid] < b) lo = mid + 1; else hi = mid;
  }
  bounds[b] = lo;
}

// ---------------------------------------------------------------------------
// Pass 1: S = Q * K^T over union edge range; mask by segment; write raw masked
// scores into final attn slots; per-row online (m, l) -> ml partials.
__global__ __launch_bounds__(THREADS) void gat_scores(
    const float* __restrict__ Q, const float* __restrict__ Km,
    const int* __restrict__ batch, float* __restrict__ attn,
    const int* __restrict__ bounds, float* __restrict__ mlOut)
{
  const int c = blockIdx.x, bt = blockIdx.y, h = blockIdx.z;
  const int lane = threadIdx.x & 31, wave = threadIdx.x >> 5;
  const int half = lane >> 4, lr = lane & 15;
  const int btbase = bt * BT;

  __shared__ float ldsQ[16 * 68];          // padded: row stride 68 floats (272B, 16B-aligned)
  __shared__ float ldsK[NW][16 * 68];
  __shared__ float mlRed[NW][BT][2];

  { // stage Q tile (16 rows x 64) cooperatively, 1 v4f per thread
    int t = threadIdx.x;
    int row = t >> 4, col = (t & 15) * 4;
    v4f q = *(const v4f*)(Q + ((long)h * BSZ + btbase + row) * DK + col);
    *(v4f*)&ldsQ[row * 68 + col] = q;
  }
  __syncthreads();

  int estart = bounds[btbase] & ~15;
  int eend = (bounds[btbase + BT] + 15) & ~15;
  if (eend > E) eend = E;
  int ntiles = (eend - estart) >> 4;
  int span = (ntiles + CHUNKS - 1) / CHUNKS;
  int t0 = c * span;
  int t1 = t0 + span; if (t1 > ntiles) t1 = ntiles;

  float m[8], l[8];
#pragma unroll
  for (int r = 0; r < 8; r++) { m[r] = NEG_INF; l[r] = 0.f; }

  const int off = 2 * half;
  float* lk = ldsK[wave];

  for (int t = t0 + wave; t < t1; t += NW) {
    int e0 = estart + (t << 4);
    const float* Kt = Km + ((long)h * E + e0) * DK;
    // stage K tile (16 x 64) coalesced, NT (streamed once)
#pragma unroll
    for (int i = 0; i < 8; i++) {
      int row = 2 * i + half;
      int col = lr * 4;
      v4f kv = __builtin_nontemporal_load((const v4f*)(Kt + row * DK + col));
      *(v4f*)&lk[row * 68 + col] = kv;
    }
    __builtin_prefetch(Kt + (long)NW * 16 * DK, 0, 1);  // global_prefetch_b8 next tile

    v8f acc = {0.f,0.f,0.f,0.f,0.f,0.f,0.f,0.f};
#pragma unroll
    for (int kd = 0; kd < 16; kd++) {
      // A (Q 16x4): lanes 0-15 -> M=lr, K={4kd,4kd+1}; lanes 16-31 -> K={4kd+2,4kd+3}
      v2f a = *(const v2f*)&ldsQ[lr * 68 + 4 * kd + off];
      // B (K^T 4x16): lanes 0-15 -> k={0,1}, N=lr; lanes 16-31 -> k={2,3}
      v2f b = *(const v2f*)&lk[lr * 68 + 4 * kd + off];
      acc = __builtin_amdgcn_wmma_f32_16x16x4_f32(false, a, false, b,
                                                  (short)0, acc, false, false);
    }

    int eIdx = e0 + lr;            // this lane's edge column
    int segb = batch[eIdx];
    float* arow = attn + (long)(h * BSZ + btbase) * E + eIdx;
#pragma unroll
    for (int r = 0; r < 8; r++) {
      int rr = r + 8 * half;       // D layout: VGPR r -> M = r + 8*(lane/16)
      float s = (segb == btbase + rr) ? acc[r] : NEG_INF;
      arow[(long)rr * E] = s;      // regular TH: keep in L2 for pass 2
      if (segb == btbase + rr) {
        float mn = fmaxf(m[r], s);
        l[r] = l[r] * __expf(m[r] - mn) + __expf(s - mn);
        m[r] = mn;
      }
    }
  }

  // reduce (m,l) across the 16 lanes of each half (edges), rows stay per-slot
#pragma unroll
  for (int d = 1; d < 16; d <<= 1) {
#pragma unroll
    for (int r = 0; r < 8; r++) {
      float mo = __shfl_xor(m[r], d, 32);
      float lo = __shfl_xor(l[r], d, 32);
      float mn = fmaxf(m[r], mo);
      float ln = 0.f;
      if (l[r] > 0.f) ln += l[r] * __expf(m[r] - mn);
      if (lo   > 0.f) ln += lo   * __expf(mo   - mn);
      m[r] = mn; l[r] = ln;
    }
  }
  if (lr == 0) {
#pragma unroll
    for (int r = 0; r < 8; r++) {
      mlRed[wave][r + 8 * half][0] = m[r];
      mlRed[wave][r + 8 * half][1] = l[r];
    }
  }
  __syncthreads();
  if (wave == 0 && threadIdx.x < BT) {
    int row = threadIdx.x;
    float M = NEG_INF, L = 0.f;
    for (int w = 0; w < NW; w++) {
      float mw = mlRed[w][row][0], lw = mlRed[w][row][1];
      float mn = fmaxf(M, mw);
      float Ln = 0.f;
      if (L  > 0.f) Ln += L  * __expf(M  - mn);
      if (lw > 0.f) Ln += lw * __expf(mw - mn);
      M = mn; L = Ln;
    }
    long idx = ((long)(h * NBT + bt) * CHUNKS + c) * BT + row;
    mlOut[2 * idx] = M;
    mlOut[2 * idx + 1] = L;
  }
}

// ---------------------------------------------------------------------------
// Combine chunk (m,l) partials -> shift = m + ln(l) per (h, b)
__global__ void gat_shift(const float* __restrict__ ml, float* __restrict__ shift) {
  int i = blockIdx.x * blockDim.x + threadIdx.x;
  if (i >= H * BSZ) return;
  int h = i / BSZ, b = i % BSZ;
  int bt = b / BT, row = b % BT;
  float M = NEG_INF, L = 0.f;
  for (int c = 0; c < CHUNKS; c++) {
    long idx = ((long)(h * NBT + bt) * CHUNKS + c) * BT + row;
    float mw = ml[2 * idx], lw = ml[2 * idx + 1];
    float mn = fmaxf(M, mw);
    float Ln = 0.f;
    if (L  > 0.f) Ln += L  * __expf(M  - mn);
    if (lw > 0.f) Ln += lw * __expf(mw - mn);
    M = mn; L = Ln;
  }
  shift[i] = (L > 0.f) ? (M + __logf(L)) : __builtin_inff();
}

// ---------------------------------------------------------------------------
// Pass 2: p = exp(s - shift) (normalized) -> attn; O_partial += P * V (WMMA)
__global__ __launch_bounds__(THREADS) void gat_attn_out(
    const float* __restrict__ V, const float* __restrict__ shift,
    float* __restrict__ attn, const int* __restrict__ bounds,
    float* __restrict__ Opart)
{
  const int c = blockIdx.x, bt = blockIdx.y, h = blockIdx.z;
  const int lane = threadIdx.x & 31, wave = threadIdx.x >> 5;
  const int half = lane >> 4, lr = lane & 15;
  const int btbase = bt * BT;

  __shared__ float ldsP[NW][16 * 20];   // row stride 20 floats (80B, 16B-aligned)
  __shared__ float ldsO[BT][DK];

  float sh = shift[h * BSZ + btbase + lr];

  int estart = bounds[btbase] & ~15;
  int eend = (bounds[btbase + BT] + 15) & ~15;
  if (eend > E) eend = E;
  int ntiles = (eend - estart) >> 4;
  int span = (ntiles + CHUNKS - 1) / CHUNKS;
  int t0 = c * span;
  int t1 = t0 + span; if (t1 > ntiles) t1 = ntiles;

  v8f acc[4];
#pragma unroll
  for (int dt = 0; dt < 4; dt++) acc[dt] = (v8f){0.f,0.f,0.f,0.f,0.f,0.f,0.f,0.f};

  const int off = 2 * half;
  float* lp = ldsP[wave];

  for (int t = t0 + wave; t < t1; t += NW) {
    int e0 = estart + (t << 4);
    // read raw scores (L2-hot from pass 1), normalize, write final p, stage P
    float* arow = attn + (long)(h * BSZ + btbase + lr) * E + e0 + half * 8;
#pragma unroll
    for (int i = 0; i < 2; i++) {
      v4f s4 = *(const v4f*)(arow + i * 4);
      v4f p4;
      p4.x = __expf(s4.x - sh); p4.y = __expf(s4.y - sh);
      p4.z = __expf(s4.z - sh); p4.w = __expf(s4.w - sh);
      __builtin_nontemporal_store(p4, (v4f*)(arow + i * 4));
      *(v4f*)&lp[lr * 20 + half * 8 + i * 4] = p4;
    }
    const float* Vt = V + ((long)h * E + e0) * DK;
#pragma unroll
    for (int ke = 0; ke < 4; ke++) {
      // A (P 16x4): lanes 0-15 -> M=lr, k={4ke,4ke+1}; lanes 16-31 -> k={4ke+2,4ke+3}
      v2f a = *(const v2f*)&lp[lr * 20 + 4 * ke + off];
#pragma unroll
      for (int dt = 0; dt < 4; dt++) {
        // B (V 4x16): lanes 0-15 -> rows 4ke+{0,1}, N=lr; lanes 16-31 -> rows 4ke+{2,3}
        v2f b;
        b.x = __builtin_nontemporal_load(Vt + (4 * ke + off + 0) * DK + dt * 16 + lr);
        b.y = __builtin_nontemporal_load(Vt + (4 * ke + off + 1) * DK + dt * 16 + lr);
        acc[dt] = __builtin_amdgcn_wmma_f32_16x16x4_f32(false, a, false, b,
                                                        (short)0, acc[dt], false, false);
      }
    }
  }

  // deterministic cross-wave O reduction in LDS (fixed wave order)
  for (int i = threadIdx.x; i < BT * DK; i += THREADS) ((float*)ldsO)[i] = 0.f;
  __syncthreads();
  for (int w = 0; w < NW; w++) {
    if (wave == w) {
#pragma unroll
      for (int dt = 0; dt < 4; dt++)
#pragma unroll
        for (int r = 0; r < 8; r++)
          ldsO[r + 8 * half][dt * 16 + lr] += acc[dt][r];
    }
    __syncthreads();
  }
  long base = ((long)(h * NBT + bt) * CHUNKS + c) * BT * DK;
  for (int i = threadIdx.x; i < BT * DK; i += THREADS)
    Opart[base + i] = ((float*)ldsO)[i];
}

// ---------------------------------------------------------------------------
__global__ void gat_reduce_out(const float* __restrict__ Opart, float* __restrict__ out) {
  int bt = blockIdx.x & 3, h = blockIdx.x >> 2;
  for (int i = threadIdx.x; i < BT * DK; i += 256) {
    float s = 0.f;
    for (int c = 0; c < CHUNKS; c++)
      s += Opart[((long)(h * NBT + bt) * CHUNKS + c) * BT * DK + i];
    out[((long)h * BSZ + bt * BT) * DK + i] = s;
  }
}

// ---------------------------------------------------------------------------
extern "C" void kernel_launch(void* const* d_in, const int* in_sizes, int n_in,
                              void* d_out, int out_size, void* d_ws, size_t ws_size,
                              hipStream_t stream) {
  const float* Q = (const float*)d_in[0];
  const float* K = (const float*)d_in[1];
  const float* V = (const float*)d_in[2];
  const int* batch = (const int*)d_in[3];  // int32 (JAX x64-off downcasts int64)
  float* out = (float*)d_out;
  float* attn = out + (long)H * BSZ * DK;

  char* ws = (char*)d_ws;
  int* bounds = (int*)ws;
  float* ml = (float*)(ws + WS_ML);
  float* shift = (float*)(ws + WS_SHIFT);
  float* Opart = (float*)(ws + WS_OPART);

  gat_zero_attn<<<8192, 256, 0, stream>>>(attn, (long)H * BSZ * E / 4);
  gat_bounds<<<1, 96, 0, stream>>>(batch, bounds);
  gat_scores<<<dim3(CHUNKS, NBT, H), THREADS, 0, stream>>>(Q, K, batch, attn, bounds, ml);
  gat_shift<<<1, 1024, 0, stream>>>(ml, shift);
  gat_attn_out<<<dim3(CHUNKS, NBT, H), THREADS, 0, stream>>>(V, shift, attn, bounds, Opart);
  gat_reduce_out<<<64, 256, 0, stream>>>(Opart, out);
}